// EdgeBlock_1855425872039
// MI455X (gfx1250) — compile-verified
//
#include <hip/hip_runtime.h>

typedef float v2f __attribute__((ext_vector_type(2)));
typedef float v8f __attribute__((ext_vector_type(8)));

#define NODE_DIM   32
#define EDGE_DIM   32
#define GLOBAL_DIM 16
#define HIDDEN     64
#define IN_DIM     112
#define KT1 28   // IN_DIM / 4  K-steps, layer 1
#define KT2 16   // HIDDEN / 4  K-steps, layer 2
#define NT1 4    // HIDDEN / 16 N-tiles, layer 1
#define NT2 2    // EDGE_DIM / 16 N-tiles, layer 2
#define WAVES 4
#define H_STRIDE 68   // 16-row h tile, stride padded for conflict-free b64 LDS access

__global__ __launch_bounds__(WAVES * 32) void edge_block_kernel(
    const float* __restrict__ nodes,
    const float* __restrict__ edges,
    const float* __restrict__ gglob,
    const int*   __restrict__ send,
    const int*   __restrict__ recv,
    const int*   __restrict__ batch,
    const float* __restrict__ W1,
    const float* __restrict__ b1,
    const float* __restrict__ W2,
    const float* __restrict__ b2,
    float* __restrict__ out,
    int E)
{
    // Weights stored pre-swizzled into WMMA B-fragment order:
    //   frag f, lane L, reg j  ->  lds[f*64 + L*2 + j] = W[k][n]
    //   with k = 4*kt + 2*(L>>4) + j,  n = 16*nt + (L&15)
    __shared__ float lds_w1[KT1 * NT1 * 64];          // 7168 floats
    __shared__ float lds_w2[KT2 * NT2 * 64];          // 2048 floats
    __shared__ float lds_h[WAVES][16 * H_STRIDE];     // per-wave h tile

    const int tid = threadIdx.x;

    for (int idx = tid; idx < KT1 * NT1 * 64; idx += blockDim.x) {
        int frag = idx >> 6;
        int kt = frag >> 2, nt = frag & 3;
        int l = (idx & 63) >> 1, j = idx & 1;
        int k = kt * 4 + 2 * (l >> 4) + j;
        int n = nt * 16 + (l & 15);
        lds_w1[idx] = W1[k * HIDDEN + n];
    }
    for (int idx = tid; idx < KT2 * NT2 * 64; idx += blockDim.x) {
        int frag = idx >> 6;
        int kt = frag >> 1, nt = frag & 1;
        int l = (idx & 63) >> 1, j = idx & 1;
        int k = kt * 4 + 2 * (l >> 4) + j;
        int n = nt * 16 + (l & 15);
        lds_w2[idx] = W2[k * EDGE_DIM + n];
    }
    __syncthreads();

    const int lane = tid & 31;
    const int wave = tid >> 5;
    const int lo = lane & 15;   // row within 16-edge tile (A) / column (B,C,D)
    const int hi = lane >> 4;   // K sub-select (A,B) / +8 row select (C,D)

    float bias1[NT1], bias2[NT2];
#pragma unroll
    for (int t = 0; t < NT1; ++t) bias1[t] = b1[t * 16 + lo];
#pragma unroll
    for (int t = 0; t < NT2; ++t) bias2[t] = b2[t * 16 + lo];

    float* hbuf = &lds_h[wave][0];

    const int nTiles = (E + 15) >> 4;
    const int waveStride = gridDim.x * WAVES;
    for (int tile = blockIdx.x * WAVES + wave; tile < nTiles; tile += waveStride) {
        int e = tile * 16 + lo;
        if (e >= E) e = E - 1;   // clamp (tail-safe; harmless duplicate row)
        const float* pe = edges + (long long)e        * EDGE_DIM;
        const float* ps = nodes + (long long)send[e]  * NODE_DIM;
        const float* pr = nodes + (long long)recv[e]  * NODE_DIM;
        const float* pg = gglob + (long long)batch[e] * GLOBAL_DIM;

        // ---- layer 1: h = relu(x @ W1 + b1),  x = [edges|src|dst|u] ----
        v8f acc[NT1];
#pragma unroll
        for (int t = 0; t < NT1; ++t)
#pragma unroll
            for (int i = 0; i < 8; ++i) acc[t][i] = bias1[t];

#pragma unroll
        for (int kt = 0; kt < KT1; ++kt) {
            const int k0 = kt * 4;
            const float* p; int off;
            if      (k0 < EDGE_DIM)                { p = pe; off = k0; }
            else if (k0 < EDGE_DIM + NODE_DIM)     { p = ps; off = k0 - EDGE_DIM; }
            else if (k0 < EDGE_DIM + 2 * NODE_DIM) { p = pr; off = k0 - EDGE_DIM - NODE_DIM; }
            else                                   { p = pg; off = k0 - EDGE_DIM - 2 * NODE_DIM; }
            v2f a = *reinterpret_cast<const v2f*>(p + off + 2 * hi);
#pragma unroll
            for (int nt = 0; nt < NT1; ++nt) {
                v2f b = *reinterpret_cast<const v2f*>(&lds_w1[(kt * NT1 + nt) * 64 + lane * 2]);
                acc[nt] = __builtin_amdgcn_wmma_f32_16x16x4_f32(
                    false, a, false, b, (short)0, acc[nt], false, false);
            }
        }

        // relu + deposit h tile into LDS (D-layout -> row-major [16][64])
#pragma unroll
        for (int t = 0; t < NT1; ++t)
#pragma unroll
            for (int r = 0; r < 8; ++r) {
                int m = r + 8 * hi;
                hbuf[m * H_STRIDE + t * 16 + lo] = fmaxf(acc[t][r], 0.0f);
            }

        // ---- layer 2: out = h @ W2 + b2 ----
        v8f acc2[NT2];
#pragma unroll
        for (int t = 0; t < NT2; ++t)
#pragma unroll
            for (int i = 0; i < 8; ++i) acc2[t][i] = bias2[t];

#pragma unroll
        for (int kt = 0; kt < KT2; ++kt) {
            v2f a = *reinterpret_cast<const v2f*>(&hbuf[lo * H_STRIDE + kt * 4 + 2 * hi]);
#pragma unroll
            for (int nt = 0; nt < NT2; ++nt) {
                v2f b = *reinterpret_cast<const v2f*>(&lds_w2[(kt * NT2 + nt) * 64 + lane * 2]);
                acc2[nt] = __builtin_amdgcn_wmma_f32_16x16x4_f32(
                    false, a, false, b, (short)0, acc2[nt], false, false);
            }
        }

        // store: lane writes (row = r + 8*hi, col = 16*t + lo) -> 64B contiguous per half-wave
#pragma unroll
        for (int t = 0; t < NT2; ++t)
#pragma unroll
            for (int r = 0; r < 8; ++r) {
                int erow = tile * 16 + r + 8 * hi;
                if (erow < E)
                    out[(long long)erow * EDGE_DIM + t * 16 + lo] = acc2[t][r];
            }
    }
}

extern "C" void kernel_launch(void* const* d_in, const int* in_sizes, int n_in,
                              void* d_out, int out_size, void* d_ws, size_t ws_size,
                              hipStream_t stream) {
    const float* nodes = (const float*)d_in[0];
    const float* edges = (const float*)d_in[1];
    const float* gglob = (const float*)d_in[2];
    const int*   send  = (const int*)d_in[3];
    const int*   recv  = (const int*)d_in[4];
    const int*   batch = (const int*)d_in[5];
    const float* W1    = (const float*)d_in[6];
    const float* b1    = (const float*)d_in[7];
    const float* W2    = (const float*)d_in[8];
    const float* b2    = (const float*)d_in[9];
    float* out = (float*)d_out;

    int E = in_sizes[3];                 // send has one entry per edge
    int nTiles = (E + 15) / 16;
    int blocks = (nTiles + WAVES - 1) / WAVES;
    if (blocks > 8192) blocks = 8192;    // persistent waves, grid-stride over tiles

    edge_block_kernel<<<blocks, WAVES * 32, 0, stream>>>(
        nodes, edges, gglob, send, recv, batch, W1, b1, W2, b2, out, E);
}